// ContrastiveLoss_53901839565361
// MI455X (gfx1250) — compile-verified
//
#include <hip/hip_runtime.h>

typedef __attribute__((ext_vector_type(2))) float v2f;
typedef __attribute__((ext_vector_type(8))) float v8f;

#define BB      8192
#define DD      128
#define NTILES  512     // 8192 / 16 column tiles

__device__ __forceinline__ float rmax16(float v) {
#pragma unroll
    for (int o = 8; o > 0; o >>= 1) v = fmaxf(v, __shfl_xor(v, o, 16));
    return v;   // replicated across each 16-lane half
}
__device__ __forceinline__ float rsum16(float v) {
#pragma unroll
    for (int o = 8; o > 0; o >>= 1) v += __shfl_xor(v, o, 16);
    return v;
}

// One 16x16 f32 tile of num = (vi/TEMP) * vj^T via 32 f32 WMMAs (K=128).
// B fragments are buffered in a full 32-entry register array so the loads can
// pipeline across the WMMA tail of the previous tile; 4 accumulator chains for ILP.
__device__ __forceinline__ v8f tile_mm(const v2f* __restrict__ A,
                                       const float* __restrict__ brow, int h) {
    v2f Bv[32];
#pragma unroll
    for (int kk = 0; kk < 32; ++kk)
        Bv[kk] = *(const v2f*)(brow + kk * 4 + h * 2);

    v8f c0 = {}, c1 = {}, c2 = {}, c3 = {};
#pragma unroll
    for (int kk = 0; kk < 32; kk += 4) {
        c0 = __builtin_amdgcn_wmma_f32_16x16x4_f32(false, A[kk + 0], false, Bv[kk + 0],
                                                   (short)0, c0, false, false);
        c1 = __builtin_amdgcn_wmma_f32_16x16x4_f32(false, A[kk + 1], false, Bv[kk + 1],
                                                   (short)0, c1, false, false);
        c2 = __builtin_amdgcn_wmma_f32_16x16x4_f32(false, A[kk + 2], false, Bv[kk + 2],
                                                   (short)0, c2, false, false);
        c3 = __builtin_amdgcn_wmma_f32_16x16x4_f32(false, A[kk + 3], false, Bv[kk + 3],
                                                   (short)0, c3, false, false);
    }
    return (c0 + c1) + (c2 + c3);
}

__global__ void supcon_zero(float* out) { out[0] = 0.0f; }

__global__ __launch_bounds__(256)
void supcon_kernel(const float* __restrict__ x,
                   const long long* __restrict__ labels,
                   float* __restrict__ out) {
    const int lane = threadIdx.x & 31;
    const int wid  = threadIdx.x >> 5;
    const int rowbase = (blockIdx.x * 8 + wid) * 16;   // this wave's 16-row band
    const int rowtile = rowbase >> 4;                  // tile holding the diagonal
    const int n = lane & 15;                           // column-in-tile / A row
    const int h = lane >> 4;                           // K-half selector

    // vi = x[:,1,:] rows (A operand): preload once, scaled by 1/TEMP (=2).
    // A lane layout for 16x16x4 f32: lane n holds M=n; h selects K={2h,2h+1} of each 4.
    const float* arow = x + ((size_t)(rowbase + n) * 2 + 1) * DD;
    v2f A[32];
#pragma unroll
    for (int kk = 0; kk < 32; ++kk) {
        v2f a = *(const v2f*)(arow + kk * 4 + h * 2);
        A[kk] = a * 2.0f;                              // fold 1/TEMP into GEMM
    }

    // Row labels (< 100, narrow to int): C/D VGPR v holds row M = v + 8*h.
    int trow[8];
#pragma unroll
    for (int v = 0; v < 8; ++v)
        trow[v] = (int)labels[(size_t)(rowbase + v + 8 * h) * 2 + 1];

    const float NEG_INF = -__builtin_inff();
    float m[8], sneg[8], mp[8], spos[8], sump[8], cntp[8], dg[8];
#pragma unroll
    for (int v = 0; v < 8; ++v) {
        m[v] = NEG_INF; sneg[v] = 0.f; mp[v] = NEG_INF; spos[v] = 0.f;
        sump[v] = 0.f; cntp[v] = 0.f; dg[v] = 0.f;
    }

    // ---------------- Pass 1: per-lane online row stats ----------------
    for (int t = 0; t < NTILES; ++t) {
        const int colbase = t * 16;
        const float* brow = x + (size_t)(colbase + n) * 2 * DD;  // vj = x[:,0,:]
        if (t + 2 < NTILES)
            __builtin_prefetch(brow + 2 * 16 * 2 * DD, 0, 0);    // global_prefetch_b8
        const int tcol = (int)labels[(size_t)(colbase + n) * 2 + 1];

        v8f c = tile_mm(A, brow, h);

        if (t == rowtile) {                    // wave-uniform: diagonal tile only
#pragma unroll
            for (int v = 0; v < 8; ++v)
                dg[v] = (n == v + 8 * h) ? c[v] : 0.f;
        }

#pragma unroll
        for (int v = 0; v < 8; ++v) {
            const float xv  = c[v];                        // already /TEMP
            const bool pos  = (trow[v] == tcol);
            // global max + neg-sum (rescaled online)
            const float nm  = fmaxf(m[v], xv);
            sneg[v] = sneg[v] * __expf(m[v] - nm) + (pos ? 0.f : __expf(xv - nm));
            m[v] = nm;
            // positive-only logsumexp (online), NaN-guarded while empty
            const float xp  = pos ? xv : NEG_INF;
            const float nmp = fmaxf(mp[v], xp);
            const float scp = (nmp == NEG_INF) ? 0.f : __expf(mp[v] - nmp);
            spos[v] = spos[v] * scp + (pos ? __expf(xv - nmp) : 0.f);
            mp[v] = nmp;
            sump[v] += pos ? xv : 0.f;
            cntp[v] += pos ? 1.f : 0.f;
        }
    }

    // ---------------- Merge across the 16-lane half ----------------
    float Mfin[8], SNfin[8], invc[8], base[8];
#pragma unroll
    for (int v = 0; v < 8; ++v) {
        const float M  = rmax16(m[v]);
        const float SN = rsum16(sneg[v] * __expf(m[v] - M));
        const float MP = rmax16(mp[v]);
        const float sc = (mp[v] == NEG_INF) ? 0.f : __expf(mp[v] - MP);
        const float SP = rsum16(spos[v] * sc);
        const float SU = rsum16(sump[v]);
        const float CN = rsum16(cntp[v]);
        const float DG = rsum16(dg[v]);
        Mfin[v] = M;
        SNfin[v] = SN;
        const float ic = 1.0f / CN;                        // >=1: self is positive
        invc[v] = ic;
        // (numerator + cnt*M)/cnt  +  spread term; pass-2 adds DLOG/cnt
        base[v] = (CN * M - SU) * ic + (MP + __logf(SP) - DG);
    }

    // ------------- Pass 2: sum_{pos j} log(exp(num - M) + negsum) -------------
    float dsum[8];
#pragma unroll
    for (int v = 0; v < 8; ++v) dsum[v] = 0.f;

    for (int t = 0; t < NTILES; ++t) {
        const int colbase = t * 16;
        const float* brow = x + (size_t)(colbase + n) * 2 * DD;
        if (t + 2 < NTILES)
            __builtin_prefetch(brow + 2 * 16 * 2 * DD, 0, 0);
        const int tcol = (int)labels[(size_t)(colbase + n) * 2 + 1];

        v8f c = tile_mm(A, brow, h);

#pragma unroll
        for (int v = 0; v < 8; ++v) {
            const bool pos = (trow[v] == tcol);
            const float lv = __logf(__expf(c[v] - Mfin[v]) + SNfin[v]);
            dsum[v] += pos ? lv : 0.f;
        }
    }

    // ---------------- Finalize: a_i, then mean ----------------
    float acc = 0.f;
#pragma unroll
    for (int v = 0; v < 8; ++v) {
        const float DL = rsum16(dsum[v]);
        const float a  = (base[v] + DL * invc[v]) * 0.5f;  // /(a_lc + a_spread)
        acc += (n == v) ? a : 0.f;                          // pick each row once
    }
#pragma unroll
    for (int o = 16; o > 0; o >>= 1) acc += __shfl_xor(acc, o, 32);
    if (lane == 0) atomicAdd(out, acc * (1.0f / (float)BB));
}

extern "C" void kernel_launch(void* const* d_in, const int* in_sizes, int n_in,
                              void* d_out, int out_size, void* d_ws, size_t ws_size,
                              hipStream_t stream) {
    (void)in_sizes; (void)n_in; (void)out_size; (void)d_ws; (void)ws_size;
    const float* x = (const float*)d_in[0];
    const long long* labels = (const long long*)d_in[1];   // int64 labels
    float* out = (float*)d_out;

    supcon_zero<<<1, 1, 0, stream>>>(out);
    supcon_kernel<<<BB / (16 * 8), 256, 0, stream>>>(x, labels, out);
}